// Densifier_24635932410235
// MI455X (gfx1250) — compile-verified
//
#include <hip/hip_runtime.h>
#include <hip/hip_bf16.h>

typedef float v2f __attribute__((ext_vector_type(2)));
typedef float v4f __attribute__((ext_vector_type(4)));
typedef float v8f __attribute__((ext_vector_type(8)));

#define GRIDC   512
#define CDIM    128
#define TWO_C   256
#define ROWS    128     // rows per workgroup tile
#define THREADS 256     // 8 waves (wave32), one 16-row band each
#define HPAD    132     // 132 % 64 == 4 -> conflict-free column (A) reads
#define CATPAD  260     // 260 % 64 == 4
#define KCHUNK  64      // weight rows staged per chunk
#define WPROWS  (KCHUNK / 2)
#define WPSTRIDE 288    // 288 % 64 == 32 -> adjacent pair-rows hit disjoint bank halves

// ---------------------------------------------------------------------------
// CDNA5 async global->LDS copy (ASYNCcnt path), 16B per lane.
// ---------------------------------------------------------------------------
__device__ __forceinline__ void async_copy_b128(const float* gaddr, void* lds_ptr) {
    unsigned ldsoff = (unsigned)(uintptr_t)lds_ptr;   // low 32 bits = LDS byte offset
    asm volatile("global_load_async_to_lds_b128 %0, %1, off"
                 :: "v"(ldsoff), "v"(gaddr) : "memory");
}
__device__ __forceinline__ void wait_asynccnt0() {
    asm volatile("s_wait_asynccnt 0x0" ::: "memory");
}

// ---------------------------------------------------------------------------
// Fused 4-layer MLP. Activations + K-pair-interleaved weight chunks in LDS;
// hot loop is ds_load_b64 (A and B) + v_wmma_f32_16x16x4_f32, no repacking.
// ---------------------------------------------------------------------------
__global__ __launch_bounds__(THREADS, 1)
void densify_mlp_kernel(const float* __restrict__ feats,
                        const float* __restrict__ skips,
                        const float* __restrict__ W_up,
                        const float* __restrict__ b_up,
                        const float* __restrict__ W_block,
                        const float* __restrict__ b_block,
                        const float* __restrict__ W_cls,
                        const float* __restrict__ b_cls,
                        float* __restrict__ out_logits,   // (L, N)
                        float* __restrict__ out_h,        // (N, C)
                        int N, int NLAYERS)
{
    __shared__ float H[ROWS][HPAD];             // h / blk tile           (~66 KB)
    __shared__ float CAT[ROWS][CATPAD];         // [u | skip] tile        (~130 KB)
    __shared__ float WP[WPROWS][WPSTRIDE];      // interleaved weights    (~36 KB)
    __shared__ float keepf[ROWS];

    const int tid   = threadIdx.x;
    const int lane  = tid & 31;
    const int wave  = tid >> 5;                 // 0..7 -> 16-row band
    const int lan16 = lane & 15;
    const int half  = lane >> 4;                // A/B lane-half select
    const size_t rowbase = (size_t)blockIdx.x * ROWS;
    const int arow  = wave * 16 + lan16;        // A-matrix row for this lane

    // ---- load feats tile into H ----
    for (int i = tid; i < ROWS * (CDIM / 4); i += THREADS) {
        int r  = i >> 5;
        int c4 = i & 31;
        ((v4f*)&H[r][0])[c4] = ((const v4f*)(feats + (rowbase + r) * CDIM))[c4];
    }
    __syncthreads();

    for (int l = 0; l < NLAYERS; ++l) {
        const float* __restrict__ Wu = W_up    + (size_t)l * CDIM * CDIM;
        const float* __restrict__ bu = b_up    + (size_t)l * CDIM;
        const float* __restrict__ Wb = W_block + (size_t)l * TWO_C * CDIM;
        const float* __restrict__ bb = b_block + (size_t)l * CDIM;
        const float* __restrict__ wc = W_cls   + (size_t)l * CDIM;
        const float  bc = b_cls[l];

        // ---- async-stage skips tile into CAT[:, 128:256] (overlaps up-GEMM) ----
        const float* __restrict__ sk = skips + ((size_t)l * N + rowbase) * CDIM;
        for (int i = tid; i < ROWS * (CDIM / 4); i += THREADS) {
            int r  = i >> 5;
            int c4 = i & 31;
            async_copy_b128(sk + (size_t)r * CDIM + 4 * c4, &CAT[r][CDIM + 4 * c4]);
        }
        __builtin_prefetch(Wb, 0, 1);   // gfx1250 global_prefetch_b8

        // ===================== u = relu(H @ Wu + bu) -> CAT[:,0:128] ==========
        {
            v8f acc[8];
            #pragma unroll
            for (int t = 0; t < 8; ++t) {
                const float bias = bu[t * 16 + lan16];
                #pragma unroll
                for (int j = 0; j < 8; ++j) acc[t][j] = bias;
            }
            for (int kb = 0; kb < CDIM; kb += KCHUNK) {
                // stage KCHUNK x 128 weight rows, K-pair interleaved:
                // WP[k/2][2n+0] = W[k][n], WP[k/2][2n+1] = W[k+1][n]
                for (int i = tid; i < WPROWS * (CDIM / 4); i += THREADS) {
                    int kk = i >> 5;
                    int c4 = i & 31;
                    v4f r0 = ((const v4f*)(Wu + (size_t)(kb + 2 * kk    ) * CDIM))[c4];
                    v4f r1 = ((const v4f*)(Wu + (size_t)(kb + 2 * kk + 1) * CDIM))[c4];
                    v2f* dst = (v2f*)&WP[kk][8 * c4];
                    dst[0] = (v2f){r0.x, r1.x};
                    dst[1] = (v2f){r0.y, r1.y};
                    dst[2] = (v2f){r0.z, r1.z};
                    dst[3] = (v2f){r0.w, r1.w};
                }
                __syncthreads();
                #pragma unroll 4
                for (int k0 = 0; k0 < KCHUNK; k0 += 4) {
                    v2f a;   // A 16x4 fp32: lanes 0-15 K=k0,k0+1; lanes 16-31 K=k0+2,k0+3
                    a.x = H[arow][kb + k0 + 2 * half];
                    a.y = H[arow][kb + k0 + 2 * half + 1];
                    const v2f* bp = (const v2f*)&WP[(k0 >> 1) + half][0];
                    #pragma unroll
                    for (int t = 0; t < 8; ++t) {
                        v2f b = bp[t * 16 + lan16];   // single ds_load_b64
                        acc[t] = __builtin_amdgcn_wmma_f32_16x16x4_f32(
                                     false, a, false, b, (short)0, acc[t], false, false);
                    }
                }
                __syncthreads();
            }
            #pragma unroll
            for (int t = 0; t < 8; ++t)
                #pragma unroll
                for (int j = 0; j < 8; ++j)
                    CAT[wave * 16 + 8 * half + j][t * 16 + lan16] = fmaxf(acc[t][j], 0.0f);
        }
        wait_asynccnt0();            // skips tile resident before block GEMM
        __syncthreads();

        // ===================== blk = relu(CAT @ Wb + bb) -> H ================
        {
            v8f acc[8];
            #pragma unroll
            for (int t = 0; t < 8; ++t) {
                const float bias = bb[t * 16 + lan16];
                #pragma unroll
                for (int j = 0; j < 8; ++j) acc[t][j] = bias;
            }
            for (int kb = 0; kb < TWO_C; kb += KCHUNK) {
                for (int i = tid; i < WPROWS * (CDIM / 4); i += THREADS) {
                    int kk = i >> 5;
                    int c4 = i & 31;
                    v4f r0 = ((const v4f*)(Wb + (size_t)(kb + 2 * kk    ) * CDIM))[c4];
                    v4f r1 = ((const v4f*)(Wb + (size_t)(kb + 2 * kk + 1) * CDIM))[c4];
                    v2f* dst = (v2f*)&WP[kk][8 * c4];
                    dst[0] = (v2f){r0.x, r1.x};
                    dst[1] = (v2f){r0.y, r1.y};
                    dst[2] = (v2f){r0.z, r1.z};
                    dst[3] = (v2f){r0.w, r1.w};
                }
                __syncthreads();
                #pragma unroll 4
                for (int k0 = 0; k0 < KCHUNK; k0 += 4) {
                    v2f a;
                    a.x = CAT[arow][kb + k0 + 2 * half];
                    a.y = CAT[arow][kb + k0 + 2 * half + 1];
                    const v2f* bp = (const v2f*)&WP[(k0 >> 1) + half][0];
                    #pragma unroll
                    for (int t = 0; t < 8; ++t) {
                        v2f b = bp[t * 16 + lan16];
                        acc[t] = __builtin_amdgcn_wmma_f32_16x16x4_f32(
                                     false, a, false, b, (short)0, acc[t], false, false);
                    }
                }
                __syncthreads();
            }
            #pragma unroll
            for (int t = 0; t < 8; ++t)
                #pragma unroll
                for (int j = 0; j < 8; ++j)
                    H[wave * 16 + 8 * half + j][t * 16 + lan16] = fmaxf(acc[t][j], 0.0f);
        }
        __syncthreads();

        // ---- logits = blk @ wc + bc; keep = logits > 0 ----
        if (tid < ROWS) {
            float dot = bc;
            #pragma unroll 8
            for (int c4 = 0; c4 < CDIM / 4; ++c4) {
                v4f hv = ((const v4f*)&H[tid][0])[c4];
                dot += hv.x * wc[4 * c4]     + hv.y * wc[4 * c4 + 1]
                     + hv.z * wc[4 * c4 + 2] + hv.w * wc[4 * c4 + 3];
            }
            out_logits[(size_t)l * N + rowbase + tid] = dot;
            keepf[tid] = dot > 0.0f ? 1.0f : 0.0f;
        }
        __syncthreads();

        // ---- h = blk * keep (in place) ----
        for (int i = tid; i < ROWS * CDIM; i += THREADS) {
            int r = i >> 7;
            int c = i & (CDIM - 1);
            H[r][c] *= keepf[r];
        }
        __syncthreads();
    }

    // ---- write final h ----
    for (int i = tid; i < ROWS * (CDIM / 4); i += THREADS) {
        int r  = i >> 5;
        int c4 = i & 31;
        ((v4f*)(out_h + (rowbase + r) * CDIM))[c4] = ((const v4f*)&H[r][0])[c4];
    }
}

// ---------------------------------------------------------------------------
// Targets: exact set membership of packed 27-bit coord keys via a 16 MiB
// bitmap in workspace (same semantics as sorted searchsorted + equality).
// ---------------------------------------------------------------------------
__device__ __forceinline__ unsigned int pack3(const int* __restrict__ p, int i) {
    return ((unsigned)p[3 * i] * GRIDC + (unsigned)p[3 * i + 1]) * GRIDC
           + (unsigned)p[3 * i + 2];
}

__global__ void zero_bitmap_kernel(unsigned int* bm, int nwords) {
    for (int i = blockIdx.x * blockDim.x + threadIdx.x; i < nwords;
         i += gridDim.x * blockDim.x)
        bm[i] = 0u;
}

__global__ void set_bits_kernel(const int* __restrict__ gt, int M, unsigned int* bm) {
    int i = blockIdx.x * blockDim.x + threadIdx.x;
    if (i < M) {
        unsigned int k = pack3(gt, i);
        atomicOr(&bm[k >> 5], 1u << (k & 31));
    }
}

__global__ void targets_kernel(const int* __restrict__ coords, int N, int L,
                               const unsigned int* __restrict__ bm,
                               float* __restrict__ out_t) {
    int i = blockIdx.x * blockDim.x + threadIdx.x;
    if (i < N) {
        unsigned int k = pack3(coords, i);
        float hit = ((bm[k >> 5] >> (k & 31)) & 1u) ? 1.0f : 0.0f;
        for (int l = 0; l < L; ++l) out_t[(size_t)l * N + i] = hit;
    }
}

__global__ void targets_brute_kernel(const int* __restrict__ coords, int N,
                                     const int* __restrict__ gt, int M, int L,
                                     float* __restrict__ out_t) {
    int i = blockIdx.x * blockDim.x + threadIdx.x;
    if (i < N) {
        unsigned int k = pack3(coords, i);
        float hit = 0.0f;
        for (int j = 0; j < M; ++j)
            if (pack3(gt, j) == k) { hit = 1.0f; break; }
        for (int l = 0; l < L; ++l) out_t[(size_t)l * N + i] = hit;
    }
}

// ---------------------------------------------------------------------------
extern "C" void kernel_launch(void* const* d_in, const int* in_sizes, int n_in,
                              void* d_out, int out_size, void* d_ws, size_t ws_size,
                              hipStream_t stream) {
    const float* feats  = (const float*)d_in[0];
    const float* skips  = (const float*)d_in[1];
    const float* W_up   = (const float*)d_in[2];
    const float* b_up   = (const float*)d_in[3];
    const float* W_blk  = (const float*)d_in[4];
    const float* b_blk  = (const float*)d_in[5];
    const float* W_cls  = (const float*)d_in[6];
    const float* b_cls  = (const float*)d_in[7];
    const int*   coords = (const int*)d_in[8];
    const int*   gt     = (const int*)d_in[9];

    const int N = in_sizes[8] / 3;                 // 131072
    const int M = in_sizes[9] / 3;                 // 100000
    const int C = in_sizes[0] / N;                 // 128
    const int L = (int)((long long)in_sizes[1] / ((long long)N * C)); // 4
    (void)C; (void)out_size; (void)n_in;

    float* out_logits  = (float*)d_out;
    float* out_targets = out_logits + (size_t)L * N;
    float* out_h       = out_targets + (size_t)L * N;

    const int nwords = (GRIDC * GRIDC * GRIDC) / 32;   // 2^27 bits
    if (ws_size >= (size_t)nwords * sizeof(unsigned int)) {
        unsigned int* bm = (unsigned int*)d_ws;
        zero_bitmap_kernel<<<2048, 256, 0, stream>>>(bm, nwords);
        set_bits_kernel<<<(M + 255) / 256, 256, 0, stream>>>(gt, M, bm);
        targets_kernel<<<(N + 255) / 256, 256, 0, stream>>>(coords, N, L, bm, out_targets);
    } else {
        targets_brute_kernel<<<(N + 255) / 256, 256, 0, stream>>>(coords, N, gt, M, L, out_targets);
    }

    densify_mlp_kernel<<<N / ROWS, THREADS, 0, stream>>>(
        feats, skips, W_up, b_up, W_blk, b_blk, W_cls, b_cls,
        out_logits, out_h, N, L);
}